// Transe_90314572300924
// MI455X (gfx1250) — compile-verified
//
#include <hip/hip_runtime.h>
#include <math.h>

typedef float v2f __attribute__((ext_vector_type(2)));
typedef float v8f __attribute__((ext_vector_type(8)));

constexpr float kMargin = 9.0f;
constexpr int kB = 32;        // batch
constexpr int kD = 64;        // embedding dim
constexpr int kE = 100000;    // num entities
constexpr int kTiles = kE / 16;        // 6250 exact
constexpr int kWavesPerBlock = 8;      // 256 threads

// ---------------------------------------------------------------------------
// Kernel 1: q[b,:] = emb_e[sub[b],:] + emb_rel[rel[b],:]; qnorm[b] = ||q[b]||^2
// Tiny (2048 elements) — one block.
// ---------------------------------------------------------------------------
__global__ void __launch_bounds__(256) transe_prep(
    const int* __restrict__ sub, const int* __restrict__ rel,
    const float* __restrict__ emb_e, const float* __restrict__ emb_rel,
    float* __restrict__ q, float* __restrict__ qnorm)
{
    __shared__ float s[kB];
    const int t = threadIdx.x;
    if (t < kB) s[t] = 0.0f;
    __syncthreads();
    for (int i = t; i < kB * kD; i += 256) {
        const int b = i >> 6;          // i / 64
        const int d = i & (kD - 1);    // i % 64
        const float v = emb_e[(size_t)sub[b] * kD + d] +
                        emb_rel[(size_t)rel[b] * kD + d];
        q[i] = v;
        atomicAdd(&s[b], v * v);       // ds_add_f32
    }
    __syncthreads();
    if (t < kB) qnorm[t] = s[t];
}

// ---------------------------------------------------------------------------
// Kernel 2: one wave per 16-entity tile. Two 16x16 f32 accumulators (rows
// 0..15 and 16..31) built from 16 chained V_WMMA_F32_16X16X4_F32 each.
// ||e||^2 accumulated per lane from the same B-fragment loads, combined via
// a lane^16 shuffle (wave32). Epilogue: MARGIN - sqrt(qn + en - 2*dot).
// ---------------------------------------------------------------------------
__global__ void __launch_bounds__(256) transe_score(
    const float* __restrict__ emb_e,
    const float* __restrict__ q,
    const float* __restrict__ qnorm,
    float* __restrict__ out)
{
    const int lane = threadIdx.x & 31;
    const int wave = threadIdx.x >> 5;
    const int tile = blockIdx.x * kWavesPerBlock + wave;
    if (tile >= kTiles) return;        // wave-uniform: EXEC stays all-ones

    const int n  = lane & 15;          // output column within tile / A row M
    const int hi = lane >> 4;          // K-pair selector (0 -> K0,K1; 1 -> K2,K3)
    const int ent = tile * 16 + n;

    const float* __restrict__ eb = emb_e + (size_t)ent * kD;   // B column
    const float* __restrict__ qa = q + (size_t)n * kD;         // A rows 0..15
    const float* __restrict__ qb = qa + 16 * kD;               // A rows 16..31

    v8f acc0 = {0.f, 0.f, 0.f, 0.f, 0.f, 0.f, 0.f, 0.f};
    v8f acc1 = {0.f, 0.f, 0.f, 0.f, 0.f, 0.f, 0.f, 0.f};
    float esq = 0.0f;

#pragma unroll
    for (int k = 0; k < 16; ++k) {
        const int koff = k * 4 + hi * 2;
        const v2f bfrag = *(const v2f*)(eb + koff);
        const v2f a0    = *(const v2f*)(qa + koff);
        const v2f a1    = *(const v2f*)(qb + koff);
        esq += bfrag.x * bfrag.x + bfrag.y * bfrag.y;
        // 8 args: (neg_a, A, neg_b, B, c_mod, C, reuse_a, reuse_b)
        acc0 = __builtin_amdgcn_wmma_f32_16x16x4_f32(
            false, a0, false, bfrag, (short)0, acc0, false, false);
        acc1 = __builtin_amdgcn_wmma_f32_16x16x4_f32(
            false, a1, false, bfrag, (short)0, acc1, false, false);
    }

    // lanes L and L+16 each hold half of ||e||^2 for entity (L%16)
    const float esqFull = esq + __shfl_xor(esq, 16, 32);

#pragma unroll
    for (int v = 0; v < 8; ++v) {
        const int r0 = v + hi * 8;     // C/D layout: VGPR v, lane half
        const int r1 = r0 + 16;
        const float d2a = qnorm[r0] + esqFull - 2.0f * acc0[v];
        const float d2b = qnorm[r1] + esqFull - 2.0f * acc1[v];
        out[(size_t)r0 * kE + ent] = kMargin - sqrtf(fmaxf(d2a, 0.0f));
        out[(size_t)r1 * kE + ent] = kMargin - sqrtf(fmaxf(d2b, 0.0f));
    }
}

// ---------------------------------------------------------------------------
// Launch: inputs in setup_inputs() order: sub, rel, emb_e, emb_rel
// ---------------------------------------------------------------------------
extern "C" void kernel_launch(void* const* d_in, const int* in_sizes, int n_in,
                              void* d_out, int out_size, void* d_ws, size_t ws_size,
                              hipStream_t stream) {
    const int*   sub     = (const int*)d_in[0];
    const int*   rel     = (const int*)d_in[1];
    const float* emb_e   = (const float*)d_in[2];
    const float* emb_rel = (const float*)d_in[3];
    float*       out     = (float*)d_out;

    float* q     = (float*)d_ws;          // 32*64 floats
    float* qnorm = q + kB * kD;           // 32 floats

    transe_prep<<<1, 256, 0, stream>>>(sub, rel, emb_e, emb_rel, q, qnorm);

    const int blocks = (kTiles + kWavesPerBlock - 1) / kWavesPerBlock;
    transe_score<<<blocks, 256, 0, stream>>>(emb_e, q, qnorm, out);
}